// GraphAttentionLayer_35948876268458
// MI455X (gfx1250) — compile-verified
//
#include <hip/hip_runtime.h>
#include <hip/hip_bf16.h>

typedef __attribute__((ext_vector_type(2))) float v2f;
typedef __attribute__((ext_vector_type(8))) float v8f;

#define NN   50000
#define EE   600000
#define ET   (EE + NN)          // edges + self loops = 650000
#define H    4
#define C    32
#define IND  128
#define OUTD 128                // H*C
#define SLOPE 0.2f

// ---------------------------------------------------------------- utilities
__device__ __forceinline__ float atomic_max_f32(float* addr, float val) {
  // monotonic int trick: positive floats order as ints, negative as reversed uints
  if (val >= 0.0f)
    return __int_as_float(atomicMax((int*)addr, __float_as_int(val)));
  else
    return __uint_as_float(atomicMin((unsigned int*)addr, __float_as_uint(val)));
}

// ---------------------------------------------------------------- init
// grid: 25000 x 256  == NN*OUTD threads exactly
__global__ void gat_init(float* __restrict__ out, float* __restrict__ mx,
                         float* __restrict__ dn) {
  int i = blockIdx.x * blockDim.x + threadIdx.x;
  out[i] = 0.0f;
  if (i < NN * H) {
    mx[i] = -__builtin_inff();
    dn[i] = 0.0f;
  }
}

// ---------------------------------------------------------------- projection
// out[m,n] = sum_k x[m,k] * W[n,k] + b[n]     (x @ W^T + b)
// One wave computes a 16x64 output strip (4 WMMA tiles) so the A fragment is
// loaded once per K-step and fed to 4 v_wmma_f32_16x16x4_f32 ops.
// Block of 128 threads = 4 waves = {side(l/r)} x {col strip 0/1} for one
// 16-row block.  grid = (3125, 1, 1).
__global__ void __launch_bounds__(128)
gat_proj(const float* __restrict__ x,
         const float* __restrict__ Wl, const float* __restrict__ bl,
         const float* __restrict__ Wr, const float* __restrict__ br,
         float* __restrict__ xl, float* __restrict__ xr) {
  const int lane = threadIdx.x & 31;
  const int wave = threadIdx.x >> 5;      // 0..3
  const int side = wave >> 1;             // 0 -> W_l, 1 -> W_r
  const int n0   = (wave & 1) * 64;       // column strip base
  const int m0   = blockIdx.x * 16;

  const float* W   = side ? Wr : Wl;
  const float* b   = side ? br : bl;
  float*       dst = side ? xr : xl;

  const int half = lane >> 4;             // 0: lanes 0-15, 1: lanes 16-31
  const int l15  = lane & 15;
  const int arow = m0 + l15;              // A: M = lane%16

  v8f acc0 = {}, acc1 = {}, acc2 = {}, acc3 = {};
  for (int k0 = 0; k0 < IND; k0 += 4) {
    // ISA 32-bit A/B 16x4 layout: lanes 0-15 hold K=k0,k0+1; lanes 16-31 K=k0+2,+3
    const int kk = k0 + (half << 1);
    v2f a;
    a.x = x[arow * IND + kk];
    a.y = x[arow * IND + kk + 1];

    const int bbase = (n0 + l15) * IND + kk;
    v2f b0, b1, b2, b3;
    b0.x = W[bbase];                b0.y = W[bbase + 1];
    b1.x = W[bbase + 16 * IND];     b1.y = W[bbase + 16 * IND + 1];
    b2.x = W[bbase + 32 * IND];     b2.y = W[bbase + 32 * IND + 1];
    b3.x = W[bbase + 48 * IND];     b3.y = W[bbase + 48 * IND + 1];

    acc0 = __builtin_amdgcn_wmma_f32_16x16x4_f32(false, a, false, b0,
                                                 (short)0, acc0, false, false);
    acc1 = __builtin_amdgcn_wmma_f32_16x16x4_f32(false, a, false, b1,
                                                 (short)0, acc1, false, false);
    acc2 = __builtin_amdgcn_wmma_f32_16x16x4_f32(false, a, false, b2,
                                                 (short)0, acc2, false, false);
    acc3 = __builtin_amdgcn_wmma_f32_16x16x4_f32(false, a, false, b3,
                                                 (short)0, acc3, false, false);
  }

  // C/D layout: VGPR g -> rows g (lanes 0-15) and g+8 (lanes 16-31)
  v8f accs[4] = {acc0, acc1, acc2, acc3};
#pragma unroll
  for (int t = 0; t < 4; ++t) {
    const int bcol = n0 + t * 16 + l15;
    const float bb = b[bcol];
#pragma unroll
    for (int g = 0; g < 8; ++g) {
      const int row = m0 + g + (half << 3);
      dst[row * OUTD + bcol] = accs[t][g] + bb;
    }
  }
}

// ---------------------------------------------------------------- edge scores
// 1 wave per edge, lane = channel within head; wave32 reduce per head.
// grid = 81250 x 256 (8 waves/block) == ET waves exactly
__global__ void __launch_bounds__(256)
gat_score(const float* __restrict__ xl, const float* __restrict__ xr,
          const float* __restrict__ att, const long long* __restrict__ ei,
          float* __restrict__ sc, float* __restrict__ mx) {
  const int lane = threadIdx.x & 31;
  const long long e = (long long)blockIdx.x * 8 + (threadIdx.x >> 5);
  long long s, d;
  if (e < EE) { s = ei[e]; d = ei[EE + e]; } else { s = d = e - EE; }

#pragma unroll
  for (int h = 0; h < H; ++h) {
    float z = xl[s * OUTD + h * C + lane] + xr[d * OUTD + h * C + lane];
    z = z > 0.0f ? z : SLOPE * z;                 // LeakyReLU
    float p = z * att[h * C + lane];
#pragma unroll
    for (int off = 16; off > 0; off >>= 1)        // wave32 sum
      p += __shfl_xor(p, off, 32);
    if (lane == 0) {
      sc[e * H + h] = p;
      atomic_max_f32(&mx[d * H + h], p);          // segment max
    }
  }
}

// ---------------------------------------------------------------- exp + segsum
__global__ void gat_expsum(float* __restrict__ sc, const float* __restrict__ mx,
                           float* __restrict__ dn, const long long* __restrict__ ei) {
  long long i = (long long)blockIdx.x * blockDim.x + threadIdx.x;
  if (i >= (long long)ET * H) return;
  long long e = i >> 2;
  int h = (int)(i & 3);
  long long d = (e < EE) ? ei[EE + e] : (e - EE);
  float w = expf(sc[i] - mx[d * H + h]);
  sc[i] = w;                                      // overwrite score with exp
  atomicAdd(&dn[d * H + h], w);
}

// ---------------------------------------------------------------- aggregate
// 1 wave per edge, lane = channel; scatter alpha * x_l[src] into out.
__global__ void __launch_bounds__(256)
gat_agg(const float* __restrict__ xl, const float* __restrict__ sc,
        const float* __restrict__ dn, const long long* __restrict__ ei,
        float* __restrict__ out) {
  const int lane = threadIdx.x & 31;
  const long long e = (long long)blockIdx.x * 8 + (threadIdx.x >> 5);
  long long s, d;
  if (e < EE) { s = ei[e]; d = ei[EE + e]; } else { s = d = e - EE; }

#pragma unroll
  for (int h = 0; h < H; ++h) {
    float alpha = sc[e * H + h] / (dn[d * H + h] + 1e-16f);
    atomicAdd(&out[d * OUTD + h * C + lane],
              alpha * xl[s * OUTD + h * C + lane]);
  }
}

// ---------------------------------------------------------------- finalize
__global__ void gat_fin(float* __restrict__ out, const float* __restrict__ bias) {
  int i = blockIdx.x * blockDim.x + threadIdx.x;  // NN*OUTD exact
  float v = out[i] + bias[i & (OUTD - 1)];
  out[i] = v > 0.0f ? v : expm1f(v);              // ELU(alpha=1)
}

// ---------------------------------------------------------------- launcher
extern "C" void kernel_launch(void* const* d_in, const int* in_sizes, int n_in,
                              void* d_out, int out_size, void* d_ws, size_t ws_size,
                              hipStream_t stream) {
  (void)in_sizes; (void)n_in; (void)out_size; (void)ws_size;

  const float*     x    = (const float*)d_in[0];
  const long long* ei   = (const long long*)d_in[1];   // int64 in reference
  const float*     Wl   = (const float*)d_in[2];
  const float*     bl   = (const float*)d_in[3];
  const float*     Wr   = (const float*)d_in[4];
  const float*     br   = (const float*)d_in[5];
  const float*     att  = (const float*)d_in[6];
  const float*     bias = (const float*)d_in[7];
  float*           out  = (float*)d_out;

  // workspace layout (floats): xl | xr | scores | max | denom  (~63 MB)
  float* xl = (float*)d_ws;
  float* xr = xl + (size_t)NN * OUTD;
  float* sc = xr + (size_t)NN * OUTD;
  float* mx = sc + (size_t)ET * H;
  float* dn = mx + (size_t)NN * H;

  gat_init<<<NN * OUTD / 256, 256, 0, stream>>>(out, mx, dn);

  gat_proj<<<NN / 16, 128, 0, stream>>>(x, Wl, bl, Wr, br, xl, xr);

  gat_score<<<ET / 8, 256, 0, stream>>>(xl, xr, att, ei, sc, mx);

  gat_expsum<<<((long long)ET * H + 255) / 256, 256, 0, stream>>>(sc, mx, dn, ei);

  gat_agg<<<ET / 8, 256, 0, stream>>>(xl, sc, dn, ei, out);

  gat_fin<<<NN * OUTD / 256, 256, 0, stream>>>(out, bias);
}